// Pointnet2Bbox_46454366274097
// MI455X (gfx1250) — compile-verified
//
#include <hip/hip_runtime.h>
#include <hip/hip_bf16.h>

// Pointnet2Bbox for MI455X (gfx1250, wave32, WMMA).
// All GEMM-like ops routed through V_WMMA_F32_16X16X4_F32 (fp32-precise).
//
// Input order assumption (setup_inputs dict insertion order, params dict
// insertion order, tuples (W, b)):
//  0 pointcloud (64,512,4)  1 semantic (64,512,64)  2 img (64,64)
//  3.. xyz_up[ (128,4)+(128), (128,128)+(128) ]
//  7.. merge [ (128,256)+(128) ]
//  9.. sa1 [ (128,131)+(128), (128,128)+(128), (128,128)+(128) ]
// 15.. sa2 [ (128,131)+(128), (128,128)+(128), (256,128)+(256) ]
// 21.. sa3 [ (256,259)+(256), (256,256)+(256), (512,256)+(512) ]
// 27.. reg [ (256,512)+(256), (256,256)+(256), (43,256)+(43) ]

typedef __attribute__((ext_vector_type(2))) float v2f;
typedef __attribute__((ext_vector_type(8))) float v8f;

__device__ __forceinline__ v8f wmma_f32(v2f a, v2f b, v8f c) {
  // D(16x16) += A(16x4) * B(4x16), fp32
  return __builtin_amdgcn_wmma_f32_16x16x4_f32(
      /*neg_a=*/false, a, /*neg_b=*/false, b,
      /*c_mod=*/(short)0, c, /*reuse_a=*/false, /*reuse_b=*/false);
}

// ---------------------------------------------------------------------------
// Stage 1: xyz_up (4->128->128) + merge([x,sem,img] 256 -> 128), all relu.
// One workgroup per (batch, 64-point tile). 256 threads = 8 waves.
// Each wave owns one 16-row M-tile, loops 4 N-tiles of 16 columns.
// ---------------------------------------------------------------------------
__global__ void stage1_kernel(const float* __restrict__ pc,
                              const float* __restrict__ sem,
                              const float* __restrict__ img,
                              const float* __restrict__ W1, const float* __restrict__ B1,
                              const float* __restrict__ W2, const float* __restrict__ B2,
                              const float* __restrict__ Wm, const float* __restrict__ Bm,
                              float* __restrict__ feat0) {
  __shared__ float sbuf[128 * 64];  // x1 then x2 (row-major [ch][col])
  __shared__ float ssem[64 * 64];
  __shared__ float simg[64];
  __shared__ float sp[4 * 64];

  const int b  = blockIdx.x >> 3;
  const int s0 = (blockIdx.x & 7) * 64;
  const int tid = threadIdx.x;

  {
    int j = tid & 63, ch = tid >> 6;  // ch 0..3
    sp[ch * 64 + j] = pc[((b * 512) + s0 + j) * 4 + ch];
  }
  for (int e = tid; e < 64 * 64; e += 256) {
    int ch = e >> 6, j = e & 63;
    ssem[e] = sem[((b * 512) + s0 + j) * 64 + ch];
  }
  if (tid < 64) simg[tid] = img[b * 64 + tid];
  __syncthreads();

  const int wave = tid >> 5, lane = tid & 31;
  const int half = lane >> 4, l15 = lane & 15;
  const int koff = half ? 2 : 0;
  const int mt = wave << 4;
  const int mrow = mt + l15;

  // ---- GEMM1: x1 = relu(W1(128x4) @ p(4x64) + b1), K=4 (one WMMA) ----
  v8f acc1[4] = {};
  {
    v2f a;
    a.x = W1[mrow * 4 + koff];
    a.y = W1[mrow * 4 + koff + 1];
    for (int nt = 0; nt < 4; ++nt) {
      int colb = nt * 16 + l15;
      v2f bb;
      bb.x = sp[koff * 64 + colb];
      bb.y = sp[(koff + 1) * 64 + colb];
      acc1[nt] = wmma_f32(a, bb, acc1[nt]);
    }
  }
  for (int nt = 0; nt < 4; ++nt)
    for (int r = 0; r < 8; ++r) {
      int row = mt + half * 8 + r;
      sbuf[row * 64 + nt * 16 + l15] = fmaxf(acc1[nt][r] + B1[row], 0.0f);
    }
  __syncthreads();

  // ---- GEMM2: x2 = relu(W2(128x128) @ x1 + b2) ----
  v8f acc2[4] = {};
  for (int k0 = 0; k0 < 128; k0 += 4) {
    int ka = k0 + koff;
    v2f a;
    a.x = W2[mrow * 128 + ka];
    a.y = W2[mrow * 128 + ka + 1];
    for (int nt = 0; nt < 4; ++nt) {
      int colb = nt * 16 + l15;
      v2f bb;
      bb.x = sbuf[ka * 64 + colb];
      bb.y = sbuf[(ka + 1) * 64 + colb];
      acc2[nt] = wmma_f32(a, bb, acc2[nt]);
    }
  }
  __syncthreads();  // all reads of x1 complete
  for (int nt = 0; nt < 4; ++nt)
    for (int r = 0; r < 8; ++r) {
      int row = mt + half * 8 + r;
      sbuf[row * 64 + nt * 16 + l15] = fmaxf(acc2[nt][r] + B2[row], 0.0f);
    }
  __syncthreads();

  // ---- GEMM3: feat = relu(Wm(128x256) @ [x2; sem; img] + bm) ----
  v8f acc3[4] = {};
  for (int k0 = 0; k0 < 256; k0 += 4) {
    int ka = k0 + koff;
    v2f a;
    a.x = Wm[mrow * 256 + ka];
    a.y = Wm[mrow * 256 + ka + 1];
    for (int nt = 0; nt < 4; ++nt) {
      int colb = nt * 16 + l15;
      v2f bb;
      if (ka < 128)       { bb.x = sbuf[ka * 64 + colb];        bb.y = sbuf[(ka + 1) * 64 + colb]; }
      else if (ka < 192)  { bb.x = ssem[(ka - 128) * 64 + colb]; bb.y = ssem[(ka - 127) * 64 + colb]; }
      else                { bb.x = simg[ka - 192];               bb.y = simg[ka - 191]; }
      acc3[nt] = wmma_f32(a, bb, acc3[nt]);
    }
  }
  for (int nt = 0; nt < 4; ++nt)
    for (int r = 0; r < 8; ++r) {
      int row = mt + half * 8 + r;
      int colg = s0 + nt * 16 + l15;
      feat0[(b * 128 + row) * 512 + colg] = fmaxf(acc3[nt][r] + Bm[row], 0.0f);
    }
}

// ---------------------------------------------------------------------------
// Farthest point sampling. One block per batch, blockDim == n (power of two).
// Matches jnp: dist0=1e10, start idx 0, argmax picks first occurrence.
// ---------------------------------------------------------------------------
__global__ void fps_kernel(const float* __restrict__ xyz, int xs, int n,
                           int npoint, int* __restrict__ out) {
  __shared__ float sx[512], sy[512], sz[512];
  __shared__ float sval[512];
  __shared__ int   sidx[512];
  const int b = blockIdx.x, tid = threadIdx.x;
  float px = xyz[(b * n + tid) * xs + 0];
  float py = xyz[(b * n + tid) * xs + 1];
  float pz = xyz[(b * n + tid) * xs + 2];
  sx[tid] = px; sy[tid] = py; sz[tid] = pz;
  if (tid == 0) out[b * npoint] = 0;
  float dist = 1e10f;
  int last = 0;
  __syncthreads();
  for (int i = 1; i < npoint; ++i) {
    float dx = px - sx[last], dy = py - sy[last], dz = pz - sz[last];
    dist = fminf(dist, dx * dx + dy * dy + dz * dz);
    sval[tid] = dist; sidx[tid] = tid;
    __syncthreads();
    for (int s = n >> 1; s > 0; s >>= 1) {
      if (tid < s) {
        float v1 = sval[tid], v2 = sval[tid + s];
        int i1 = sidx[tid], i2 = sidx[tid + s];
        if (v2 > v1 || (v2 == v1 && i2 < i1)) { sval[tid] = v2; sidx[tid] = i2; }
      }
      __syncthreads();
    }
    last = sidx[0];
    if (tid == 0) out[b * npoint + i] = last;
    __syncthreads();
  }
}

// ---------------------------------------------------------------------------
// Ball query: first `nsample` in-radius indices in ascending order, padded
// with the first hit (or 0 if no hits). One thread per (batch, centroid).
// Also gathers new_xyz.
// ---------------------------------------------------------------------------
__global__ void ballquery_kernel(const float* __restrict__ xyz, int xs,
                                 const int* __restrict__ fidx, int n, int np,
                                 float r2, float* __restrict__ nxyz,
                                 int* __restrict__ idx, int total) {
  int t = blockIdx.x * blockDim.x + threadIdx.x;
  if (t >= total) return;
  int b = t / np, c = t % np;
  int fi = fidx[b * np + c];
  float cx = xyz[(b * n + fi) * xs + 0];
  float cy = xyz[(b * n + fi) * xs + 1];
  float cz = xyz[(b * n + fi) * xs + 2];
  nxyz[(b * np + c) * 3 + 0] = cx;
  nxyz[(b * np + c) * 3 + 1] = cy;
  nxyz[(b * np + c) * 3 + 2] = cz;
  int* op = idx + (b * np + c) * 64;
  int cnt = 0, first = 0;
  for (int i = 0; i < n; ++i) {
    float dx = xyz[(b * n + i) * xs + 0] - cx;
    float dy = xyz[(b * n + i) * xs + 1] - cy;
    float dz = xyz[(b * n + i) * xs + 2] - cz;
    if (dx * dx + dy * dy + dz * dz < r2) {
      if (cnt == 0) first = i;
      op[cnt++] = i;
      if (cnt >= 64) break;
    }
  }
  for (int j = cnt; j < 64; ++j) op[j] = first;
}

// ---------------------------------------------------------------------------
// Fused SA MLP: gather group (131x64) to LDS, three WMMA GEMMs with relu,
// column-max via shuffles. One workgroup per (batch, centroid), 8 waves.
// Cin fixed at 128 (C1 = 131, padded to 132). C3 in {128, 256}.
// ---------------------------------------------------------------------------
__global__ void sa_mlp_kernel(const float* __restrict__ xyz, int xs,
                              const float* __restrict__ featin,
                              const float* __restrict__ nxyz,
                              const int* __restrict__ idx,
                              const float* __restrict__ W1, const float* __restrict__ B1,
                              const float* __restrict__ W2, const float* __restrict__ B2,
                              const float* __restrict__ W3, const float* __restrict__ B3,
                              float* __restrict__ featout,
                              int n, int np, int C3) {
  __shared__ float sg[132 * 64];
  __shared__ int   sidx[64];
  __shared__ float sc[3];
  const int b = blockIdx.x / np, c = blockIdx.x % np;
  const int tid = threadIdx.x;

  if (tid < 64) sidx[tid] = idx[(b * np + c) * 64 + tid];
  if (tid < 3)  sc[tid] = nxyz[(b * np + c) * 3 + tid];
  __syncthreads();
  for (int e = tid; e < 132 * 64; e += 256) {
    int row = e >> 6, j = e & 63;
    float v;
    if (row < 3)        v = xyz[(b * n + sidx[j]) * xs + row] - sc[row];
    else if (row < 131) v = featin[(b * 128 + (row - 3)) * n + sidx[j]];
    else                v = 0.0f;
    sg[e] = v;
  }
  __syncthreads();

  const int wave = tid >> 5, lane = tid & 31;
  const int half = lane >> 4, l15 = lane & 15;
  const int koff = half ? 2 : 0;
  const int mt = wave << 4;
  const int mrow = mt + l15;

  // ---- layer 1: 128x131 @ 131x64, K padded to 132 ----
  v8f acc1[4] = {};
  for (int k0 = 0; k0 < 132; k0 += 4) {
    int ka = k0 + koff;
    v2f a;
    a.x = (ka     < 131) ? W1[mrow * 131 + ka]     : 0.0f;
    a.y = (ka + 1 < 131) ? W1[mrow * 131 + ka + 1] : 0.0f;
    for (int nt = 0; nt < 4; ++nt) {
      int colb = nt * 16 + l15;
      v2f bb;
      bb.x = sg[ka * 64 + colb];
      bb.y = sg[(ka + 1) * 64 + colb];
      acc1[nt] = wmma_f32(a, bb, acc1[nt]);
    }
  }
  __syncthreads();  // all reads of grouped data done
  for (int nt = 0; nt < 4; ++nt)
    for (int r = 0; r < 8; ++r) {
      int row = mt + half * 8 + r;
      sg[row * 64 + nt * 16 + l15] = fmaxf(acc1[nt][r] + B1[row], 0.0f);
    }
  __syncthreads();

  // ---- layer 2: 128x128 @ 128x64 ----
  v8f acc2[4] = {};
  for (int k0 = 0; k0 < 128; k0 += 4) {
    int ka = k0 + koff;
    v2f a;
    a.x = W2[mrow * 128 + ka];
    a.y = W2[mrow * 128 + ka + 1];
    for (int nt = 0; nt < 4; ++nt) {
      int colb = nt * 16 + l15;
      v2f bb;
      bb.x = sg[ka * 64 + colb];
      bb.y = sg[(ka + 1) * 64 + colb];
      acc2[nt] = wmma_f32(a, bb, acc2[nt]);
    }
  }
  __syncthreads();
  for (int nt = 0; nt < 4; ++nt)
    for (int r = 0; r < 8; ++r) {
      int row = mt + half * 8 + r;
      sg[row * 64 + nt * 16 + l15] = fmaxf(acc2[nt][r] + B2[row], 0.0f);
    }
  __syncthreads();

  // ---- layer 3: C3x128 @ 128x64, then max over 64 columns ----
  const int mtiles = C3 >> 4;
  for (int mi = wave; mi < mtiles; mi += 8) {
    int mt3 = mi << 4;
    int mrow3 = mt3 + l15;
    v8f acc3[4] = {};
    for (int k0 = 0; k0 < 128; k0 += 4) {
      int ka = k0 + koff;
      v2f a;
      a.x = W3[mrow3 * 128 + ka];
      a.y = W3[mrow3 * 128 + ka + 1];
      for (int nt = 0; nt < 4; ++nt) {
        int colb = nt * 16 + l15;
        v2f bb;
        bb.x = sg[ka * 64 + colb];
        bb.y = sg[(ka + 1) * 64 + colb];
        acc3[nt] = wmma_f32(a, bb, acc3[nt]);
      }
    }
    for (int r = 0; r < 8; ++r) {
      int row = mt3 + half * 8 + r;
      float mv = fmaxf(fmaxf(acc3[0][r], acc3[1][r]), fmaxf(acc3[2][r], acc3[3][r]));
      mv = fmaxf(mv, __shfl_xor(mv, 1, 32));
      mv = fmaxf(mv, __shfl_xor(mv, 2, 32));
      mv = fmaxf(mv, __shfl_xor(mv, 4, 32));
      mv = fmaxf(mv, __shfl_xor(mv, 8, 32));
      if (l15 == 0)
        featout[(b * C3 + row) * np + c] = fmaxf(mv + B3[row], 0.0f);
    }
  }
}

// ---------------------------------------------------------------------------
// SA3 (global): g = [xyz^T(3x32); feat(256x32)] -> 256 -> 256 -> 512, relu,
// max over 32 columns. One workgroup per batch.
// ---------------------------------------------------------------------------
__global__ void sa3_kernel(const float* __restrict__ nxyz2,
                           const float* __restrict__ feat2,
                           const float* __restrict__ Wa, const float* __restrict__ Ba,
                           const float* __restrict__ Wb, const float* __restrict__ Bb,
                           const float* __restrict__ Wc, const float* __restrict__ Bc,
                           float* __restrict__ feat3) {
  __shared__ float sg[260 * 32];
  const int b = blockIdx.x, tid = threadIdx.x;
  for (int e = tid; e < 260 * 32; e += 256) {
    int row = e >> 5, j = e & 31;
    float v;
    if (row < 3)        v = nxyz2[(b * 32 + j) * 3 + row];
    else if (row < 259) v = feat2[(b * 256 + (row - 3)) * 32 + j];
    else                v = 0.0f;
    sg[e] = v;
  }
  __syncthreads();

  const int wave = tid >> 5, lane = tid & 31;
  const int half = lane >> 4, l15 = lane & 15;
  const int koff = half ? 2 : 0;

  // ---- layer 1: 256x259 @ 259x32, K padded to 260 ----
  v8f acc1[2][2] = {};
  for (int mi = 0; mi < 2; ++mi) {
    int mrow = (wave * 2 + mi) * 16 + l15;
    for (int k0 = 0; k0 < 260; k0 += 4) {
      int ka = k0 + koff;
      v2f a;
      a.x = (ka     < 259) ? Wa[mrow * 259 + ka]     : 0.0f;
      a.y = (ka + 1 < 259) ? Wa[mrow * 259 + ka + 1] : 0.0f;
      for (int nt = 0; nt < 2; ++nt) {
        int colb = nt * 16 + l15;
        v2f bb;
        bb.x = sg[ka * 32 + colb];
        bb.y = sg[(ka + 1) * 32 + colb];
        acc1[mi][nt] = wmma_f32(a, bb, acc1[mi][nt]);
      }
    }
  }
  __syncthreads();
  for (int mi = 0; mi < 2; ++mi)
    for (int nt = 0; nt < 2; ++nt)
      for (int r = 0; r < 8; ++r) {
        int row = (wave * 2 + mi) * 16 + half * 8 + r;
        sg[row * 32 + nt * 16 + l15] = fmaxf(acc1[mi][nt][r] + Ba[row], 0.0f);
      }
  __syncthreads();

  // ---- layer 2: 256x256 @ 256x32 ----
  v8f acc2[2][2] = {};
  for (int mi = 0; mi < 2; ++mi) {
    int mrow = (wave * 2 + mi) * 16 + l15;
    for (int k0 = 0; k0 < 256; k0 += 4) {
      int ka = k0 + koff;
      v2f a;
      a.x = Wb[mrow * 256 + ka];
      a.y = Wb[mrow * 256 + ka + 1];
      for (int nt = 0; nt < 2; ++nt) {
        int colb = nt * 16 + l15;
        v2f bb;
        bb.x = sg[ka * 32 + colb];
        bb.y = sg[(ka + 1) * 32 + colb];
        acc2[mi][nt] = wmma_f32(a, bb, acc2[mi][nt]);
      }
    }
  }
  __syncthreads();
  for (int mi = 0; mi < 2; ++mi)
    for (int nt = 0; nt < 2; ++nt)
      for (int r = 0; r < 8; ++r) {
        int row = (wave * 2 + mi) * 16 + half * 8 + r;
        sg[row * 32 + nt * 16 + l15] = fmaxf(acc2[mi][nt][r] + Bb[row], 0.0f);
      }
  __syncthreads();

  // ---- layer 3: 512x256 @ 256x32, then max over 32 columns ----
  for (int mi = 0; mi < 4; ++mi) {
    int mt3 = (wave * 4 + mi) * 16;
    int mrow = mt3 + l15;
    v8f acc3[2] = {};
    for (int k0 = 0; k0 < 256; k0 += 4) {
      int ka = k0 + koff;
      v2f a;
      a.x = Wc[mrow * 256 + ka];
      a.y = Wc[mrow * 256 + ka + 1];
      for (int nt = 0; nt < 2; ++nt) {
        int colb = nt * 16 + l15;
        v2f bb;
        bb.x = sg[ka * 32 + colb];
        bb.y = sg[(ka + 1) * 32 + colb];
        acc3[nt] = wmma_f32(a, bb, acc3[nt]);
      }
    }
    for (int r = 0; r < 8; ++r) {
      int row = mt3 + half * 8 + r;
      float mv = fmaxf(acc3[0][r], acc3[1][r]);
      mv = fmaxf(mv, __shfl_xor(mv, 1, 32));
      mv = fmaxf(mv, __shfl_xor(mv, 2, 32));
      mv = fmaxf(mv, __shfl_xor(mv, 4, 32));
      mv = fmaxf(mv, __shfl_xor(mv, 8, 32));
      if (l15 == 0)
        feat3[b * 512 + row] = fmaxf(mv + Bc[row], 0.0f);
    }
  }
}

// ---------------------------------------------------------------------------
// Generic batched dense layer via WMMA: out[n*M+m] = act(sum_c W[m,c]*X[n,c]+b[m]).
// X row-major (N,C); one 16x16 output tile per wave; block = 4 waves.
// ---------------------------------------------------------------------------
__global__ void gemm_nt_kernel(const float* __restrict__ W,
                               const float* __restrict__ bias,
                               const float* __restrict__ X,
                               float* __restrict__ out,
                               int M, int N, int C, int do_relu) {
  const int wave = threadIdx.x >> 5, lane = threadIdx.x & 31;
  const int half = lane >> 4, l15 = lane & 15;
  const int koff = half ? 2 : 0;
  const int ntiles = N >> 4;
  const int mtiles = (M + 15) >> 4;
  int tile = blockIdx.x * (blockDim.x >> 5) + wave;
  if (tile >= mtiles * ntiles) return;  // wave-uniform
  const int mt = (tile / ntiles) << 4;
  const int nt = (tile % ntiles) << 4;
  const int mrow = mt + l15;
  const int col = nt + l15;
  v8f acc = {};
  for (int k0 = 0; k0 < C; k0 += 4) {
    int ka = k0 + koff;
    v2f a, bb;
    a.x = (mrow < M) ? W[mrow * C + ka] : 0.0f;
    a.y = (mrow < M) ? W[mrow * C + ka + 1] : 0.0f;
    bb.x = X[col * C + ka];
    bb.y = X[col * C + ka + 1];
    acc = wmma_f32(a, bb, acc);
  }
  for (int r = 0; r < 8; ++r) {
    int row = mt + half * 8 + r;
    if (row < M) {
      float v = acc[r] + bias[row];
      if (do_relu) v = fmaxf(v, 0.0f);
      out[col * M + row] = v;
    }
  }
}

// ---------------------------------------------------------------------------
extern "C" void kernel_launch(void* const* d_in, const int* in_sizes, int n_in,
                              void* d_out, int out_size, void* d_ws, size_t ws_size,
                              hipStream_t stream) {
  (void)in_sizes; (void)n_in; (void)out_size; (void)ws_size;
  const float* pc  = (const float*)d_in[0];
  const float* sem = (const float*)d_in[1];
  const float* img = (const float*)d_in[2];
  int p = 3;
  const float* xuW1 = (const float*)d_in[p++]; const float* xuB1 = (const float*)d_in[p++];
  const float* xuW2 = (const float*)d_in[p++]; const float* xuB2 = (const float*)d_in[p++];
  const float* mW   = (const float*)d_in[p++]; const float* mB   = (const float*)d_in[p++];
  const float* s1W1 = (const float*)d_in[p++]; const float* s1B1 = (const float*)d_in[p++];
  const float* s1W2 = (const float*)d_in[p++]; const float* s1B2 = (const float*)d_in[p++];
  const float* s1W3 = (const float*)d_in[p++]; const float* s1B3 = (const float*)d_in[p++];
  const float* s2W1 = (const float*)d_in[p++]; const float* s2B1 = (const float*)d_in[p++];
  const float* s2W2 = (const float*)d_in[p++]; const float* s2B2 = (const float*)d_in[p++];
  const float* s2W3 = (const float*)d_in[p++]; const float* s2B3 = (const float*)d_in[p++];
  const float* s3W1 = (const float*)d_in[p++]; const float* s3B1 = (const float*)d_in[p++];
  const float* s3W2 = (const float*)d_in[p++]; const float* s3B2 = (const float*)d_in[p++];
  const float* s3W3 = (const float*)d_in[p++]; const float* s3B3 = (const float*)d_in[p++];
  const float* rW1  = (const float*)d_in[p++]; const float* rB1  = (const float*)d_in[p++];
  const float* rW2  = (const float*)d_in[p++]; const float* rB2  = (const float*)d_in[p++];
  const float* rW3  = (const float*)d_in[p++]; const float* rB3  = (const float*)d_in[p++];

  // Workspace layout (~26 MB)
  float* feat0 = (float*)d_ws;                 // 64*128*512
  float* feat1 = feat0 + 64 * 128 * 512;       // 64*128*128
  float* feat2 = feat1 + 64 * 128 * 128;       // 64*256*32
  float* feat3 = feat2 + 64 * 256 * 32;        // 64*512
  float* nxyz1 = feat3 + 64 * 512;             // 64*128*3
  float* nxyz2 = nxyz1 + 64 * 128 * 3;         // 64*32*3
  float* reg1  = nxyz2 + 64 * 32 * 3;          // 64*256
  float* reg2  = reg1  + 64 * 256;             // 64*256
  int* fidx1 = (int*)(reg2 + 64 * 256);        // 64*128
  int* fidx2 = fidx1 + 64 * 128;               // 64*32
  int* idx1  = fidx2 + 64 * 32;                // 64*128*64
  int* idx2  = idx1  + 64 * 128 * 64;          // 64*32*64

  // Stage 1: per-point MLP + merge -> feat0 (64,128,512)
  stage1_kernel<<<512, 256, 0, stream>>>(pc, sem, img, xuW1, xuB1, xuW2, xuB2,
                                         mW, mB, feat0);

  // SA1: npoint=128, r=0.2, nsample=64
  fps_kernel<<<64, 512, 0, stream>>>(pc, 4, 512, 128, fidx1);
  ballquery_kernel<<<(64 * 128 + 255) / 256, 256, 0, stream>>>(
      pc, 4, fidx1, 512, 128, 0.04f, nxyz1, idx1, 64 * 128);
  sa_mlp_kernel<<<64 * 128, 256, 0, stream>>>(
      pc, 4, feat0, nxyz1, idx1, s1W1, s1B1, s1W2, s1B2, s1W3, s1B3,
      feat1, 512, 128, 128);

  // SA2: npoint=32, r=0.4, nsample=64
  fps_kernel<<<64, 128, 0, stream>>>(nxyz1, 3, 128, 32, fidx2);
  ballquery_kernel<<<(64 * 32 + 255) / 256, 256, 0, stream>>>(
      nxyz1, 3, fidx2, 128, 32, 0.16f, nxyz2, idx2, 64 * 32);
  sa_mlp_kernel<<<64 * 32, 256, 0, stream>>>(
      nxyz1, 3, feat1, nxyz2, idx2, s2W1, s2B1, s2W2, s2B2, s2W3, s2B3,
      feat2, 128, 32, 256);

  // SA3 global -> feat3 (64,512)
  sa3_kernel<<<64, 256, 0, stream>>>(nxyz2, feat2, s3W1, s3B1, s3W2, s3B2,
                                     s3W3, s3B3, feat3);

  // Regression head, batched over B=64 as the GEMM N-dimension.
  gemm_nt_kernel<<<16, 128, 0, stream>>>(rW1, rB1, feat3, reg1, 256, 64, 512, 1);
  gemm_nt_kernel<<<16, 128, 0, stream>>>(rW2, rB2, reg1, reg2, 256, 64, 256, 1);
  gemm_nt_kernel<<<3, 128, 0, stream>>>(rW3, rB3, reg2, (float*)d_out, 43, 64, 256, 0);
}